// Rank_IGR_Loss_13967233647034
// MI455X (gfx1250) — compile-verified
//
#include <hip/hip_runtime.h>
#include <hip/hip_bf16.h>
#include <math.h>

typedef __attribute__((ext_vector_type(2))) float v2f;
typedef __attribute__((ext_vector_type(8))) float v8f;

#define G1C 3.0f
#define G2C 1.0f
#define BIGP 3.0e38f
#define BIGN (-3.0e38f)

// ---------------------------------------------------------------------------
// Kernel 0: per-element precompute.
// For each (b,n): prob=exp(cls[...,1]), iou, d, pos; emit
//   A1 = exp(-G1*prob)   A2 = exp(-G2*iou)
//   B1 = exp(+G1*prob)   B2 = exp(+G2*iou)
//   DI = pos ? d : +3e38   (i-role: kills row)
//   DJ = pos ? d : -3e38   (j-role: kills column)
// ---------------------------------------------------------------------------
__global__ void rank_pre_kernel(const float* __restrict__ cls,
                                const int* __restrict__ label_cls,
                                const float* __restrict__ label_loc,
                                const float* __restrict__ pred_bboxes,
                                const float* __restrict__ label_target,
                                float* __restrict__ A1, float* __restrict__ A2,
                                float* __restrict__ B1, float* __restrict__ B2,
                                float* __restrict__ DI, float* __restrict__ DJ,
                                int N, int total) {
  int idx = blockIdx.x * blockDim.x + threadIdx.x;
  if (idx >= total) return;
  int b = idx / N;
  int n = idx - b * N;

  float prob = expf(cls[idx * 2 + 1]);
  bool pos = label_cls[idx] > 0;

  float lt0 = label_target[b * 4 + 0];
  float lt1 = label_target[b * 4 + 1];
  float lt2 = label_target[b * 4 + 2];
  float lt3 = label_target[b * 4 + 3];

  const float* pb = pred_bboxes + (size_t)b * 4 * N;
  float x1 = pb[0 * N + n], y1 = pb[1 * N + n];
  float x2 = pb[2 * N + n], y2 = pb[3 * N + n];

  float ww = fmaxf(fminf(lt2, x2) - fmaxf(lt0, x1), 0.0f);
  float hh = fmaxf(fminf(lt3, y2) - fmaxf(lt1, y1), 0.0f);
  float area = (x2 - x1) * (y2 - y1);
  float ta = (lt2 - lt0) * (lt3 - lt1);
  float inter = ww * hh;
  float iou = inter / (area + ta - inter);

  const float* ll = label_loc + (size_t)b * 4 * N;
  float cx = ll[0 * N + n] + lt0;
  float cy = ll[1 * N + n] + lt1;
  float tcx = (lt0 + lt2) * 0.5f;
  float tcy = (lt1 + lt3) * 0.5f;
  float dx = cx - tcx, dy = cy - tcy;
  float d = sqrtf(dx * dx + dy * dy);

  A1[idx] = expf(-G1C * prob);
  B1[idx] = expf(G1C * prob);
  A2[idx] = expf(-G2C * iou);
  B2[idx] = expf(G2C * iou);
  DI[idx] = pos ? d : BIGP;
  DJ[idx] = pos ? d : BIGN;
}

// ---------------------------------------------------------------------------
// Kernel 1: masked bilinear reduction via V_WMMA_F32_16X16X4_F32.
// Block = 256 threads (8 wave32). Each wave owns one 16-column j-tile of one
// batch and chains 512 f32 WMMAs over the i (K=4) dimension.
//   A(16x4): row0 = exp(-G1*prob_i), row1 = exp(-G2*iou_i), row2 = 1, rest 0
//   B(4x16): mask(i,j) = (DJ[j] - DI[i] >= 1) ? 1 : 0
//   D(16x16) accumulates; rows 0/1/2 give col-sums for loss1/loss2/count.
// Per-batch DI/A1/A2 staged in LDS (24KB of 320KB/WGP).
// ---------------------------------------------------------------------------
__global__ void __launch_bounds__(256)
rank_pair_kernel(const float* __restrict__ A1, const float* __restrict__ A2,
                 const float* __restrict__ B1, const float* __restrict__ B2,
                 const float* __restrict__ DI, const float* __restrict__ DJ,
                 float* __restrict__ P1, float* __restrict__ P2,
                 float* __restrict__ PC,
                 int N, int jtilesPerBatch, int groupsPerBatch) {
  __shared__ float sDI[2048];
  __shared__ float sA1[2048];
  __shared__ float sA2[2048];

  const int b = blockIdx.x / groupsPerBatch;
  const int g = blockIdx.x - b * groupsPerBatch;
  const int wave = threadIdx.x >> 5;
  const int lane = threadIdx.x & 31;
  const int jtile = g * 8 + wave;
  const int jbase = jtile * 16;
  const int m = lane & 15;       // row of A / column of D for this lane
  const int ksel = lane >> 4;    // 0: K={0,1}, 1: K={2,3}

  const size_t base = (size_t)b * N;

  // stage per-batch vectors into LDS
  for (int t = threadIdx.x; t < N; t += 256) {
    sDI[t] = DI[base + t];
    sA1[t] = A1[base + t];
    sA2[t] = A2[base + t];
  }
  __syncthreads();

  // per-column (j) constants for this lane
  const int j = jbase + m;
  const float djv = DJ[base + j];
  const float b1j = B1[base + j];
  const float b2j = B2[base + j];

  // A-row selection without branches in the hot loop
  const float* aSrc = (m == 1) ? sA2 : sA1;
  const float flagL = (m < 2) ? 1.0f : 0.0f;   // rows 0,1: load coeff
  const float flagC = (m == 2) ? 1.0f : 0.0f;  // row 2: ones (count)

  v8f acc = {0.0f, 0.0f, 0.0f, 0.0f, 0.0f, 0.0f, 0.0f, 0.0f};

  for (int ib = 0; ib < N; ib += 4) {
    const int i0 = ib + 2 * ksel;
    const float2 dp = *(const float2*)&sDI[i0];   // ds_load_b64
    const float2 ap = *(const float2*)&aSrc[i0];  // ds_load_b64

    v2f a;
    a.x = flagL * ap.x + flagC;
    a.y = flagL * ap.y + flagC;

    v2f bm;
    bm.x = (djv - dp.x >= 1.0f) ? 1.0f : 0.0f;
    bm.y = (djv - dp.y >= 1.0f) ? 1.0f : 0.0f;

    acc = __builtin_amdgcn_wmma_f32_16x16x4_f32(
        false, a, false, bm, (short)0, acc, false, false);
  }

  // lanes 0-15: acc[r] = D[r][jbase+lane]; lanes 16-31 hold rows 8..15 == 0
  float s1 = acc[0] * b1j;
  float s2 = acc[1] * b2j;
  float sc = acc[2];

  for (int off = 16; off > 0; off >>= 1) {
    s1 += __shfl_xor(s1, off, 32);
    s2 += __shfl_xor(s2, off, 32);
    sc += __shfl_xor(sc, off, 32);
  }

  if (lane == 0) {
    const int p = b * jtilesPerBatch + jtile;
    P1[p] = s1;
    P2[p] = s2;
    PC[p] = sc;
  }
}

// ---------------------------------------------------------------------------
// Kernel 2: deterministic final reduction over per-tile partials.
// 16 threads per batch fold jtilesPerBatch partials, then thread 0 combines
// the batch losses with the valid-mask normalization. Output: (f1, f2).
// ---------------------------------------------------------------------------
__global__ void __launch_bounds__(256)
rank_final_kernel(const float* __restrict__ P1, const float* __restrict__ P2,
                  const float* __restrict__ PC,
                  const int* __restrict__ dataset_id,
                  float* __restrict__ out, int B, int jtilesPerBatch) {
  __shared__ float L1[16], L2[16], VV[16];
  const int tid = threadIdx.x;
  const int b = tid >> 4;
  const int s = tid & 15;

  float s1 = 0.0f, s2 = 0.0f, sc = 0.0f;
  if (b < B) {
    for (int t = s; t < jtilesPerBatch; t += 16) {
      const int p = b * jtilesPerBatch + t;
      s1 += P1[p];
      s2 += P2[p];
      sc += PC[p];
    }
  }
  // reduce within each aligned 16-lane group of the wave
  for (int off = 8; off > 0; off >>= 1) {
    s1 += __shfl_xor(s1, off, 32);
    s2 += __shfl_xor(s2, off, 32);
    sc += __shfl_xor(sc, off, 32);
  }
  if (b < B && s == 0) {
    float denom = fmaxf(sc, 1.0f);
    float valid = ((dataset_id[b] != 1) && (sc > 0.0f)) ? 1.0f : 0.0f;
    L1[b] = (s1 / denom) * valid;
    L2[b] = (s2 / denom) * valid;
    VV[b] = valid;
  }
  __syncthreads();
  if (tid == 0) {
    float nv = 0.0f, t1 = 0.0f, t2 = 0.0f;
    for (int i = 0; i < B; ++i) {
      nv += VV[i];
      t1 += L1[i];
      t2 += L2[i];
    }
    float safe = fmaxf(nv, 1.0f);
    out[0] = (nv > 0.0f) ? (t1 / safe) : 0.0f;
    out[1] = (nv > 0.0f) ? (t2 / safe) : 0.0f;
  }
}

extern "C" void kernel_launch(void* const* d_in, const int* in_sizes, int n_in,
                              void* d_out, int out_size, void* d_ws, size_t ws_size,
                              hipStream_t stream) {
  const float* cls          = (const float*)d_in[0];
  const int*   label_cls    = (const int*)d_in[1];
  const float* label_loc    = (const float*)d_in[2];
  const float* pred_bboxes  = (const float*)d_in[3];
  const float* label_target = (const float*)d_in[4];
  const int*   dataset_id   = (const int*)d_in[5];

  const int B = in_sizes[5];          // dataset_id: (B,)
  const int N = in_sizes[1] / B;      // label_cls: (B,N)
  const int BN = B * N;
  const int jtilesPerBatch = N / 16;      // 128
  const int groupsPerBatch = jtilesPerBatch / 8;  // 16 (8 waves/block)

  float* w = (float*)d_ws;
  float* A1 = w + 0 * (size_t)BN;
  float* A2 = w + 1 * (size_t)BN;
  float* B1 = w + 2 * (size_t)BN;
  float* B2 = w + 3 * (size_t)BN;
  float* DI = w + 4 * (size_t)BN;
  float* DJ = w + 5 * (size_t)BN;
  float* P1 = w + 6 * (size_t)BN;
  float* P2 = P1 + (size_t)B * jtilesPerBatch;
  float* PC = P2 + (size_t)B * jtilesPerBatch;

  rank_pre_kernel<<<(BN + 255) / 256, 256, 0, stream>>>(
      cls, label_cls, label_loc, pred_bboxes, label_target,
      A1, A2, B1, B2, DI, DJ, N, BN);

  rank_pair_kernel<<<B * groupsPerBatch, 256, 0, stream>>>(
      A1, A2, B1, B2, DI, DJ, P1, P2, PC, N, jtilesPerBatch, groupsPerBatch);

  rank_final_kernel<<<1, 256, 0, stream>>>(
      P1, P2, PC, dataset_id, (float*)d_out, B, jtilesPerBatch);
}